// HASeparator_78219944394856
// MI455X (gfx1250) — compile-verified
//
#include <hip/hip_runtime.h>
#include <math.h>

// Problem constants (match reference)
#define B 256
#define D 128
#define C 8000
#define MARGIN 0.5f
#define SCALE 64.0f
#define EPS 1e-12f

typedef __attribute__((ext_vector_type(2))) float v2f;
typedef __attribute__((ext_vector_type(8))) float v8f;

// ---------------------------------------------------------------------------
// Kernel 1: inverse column norms of w:  invn[c] = rsqrt(max(sum_d w[d,c]^2, eps))
// ---------------------------------------------------------------------------
__global__ __launch_bounds__(256) void wnorm_kernel(const float* __restrict__ w,
                                                    float* __restrict__ invn) {
    int c = blockIdx.x * blockDim.x + threadIdx.x;
    if (c >= C) return;
    float s = 0.f;
#pragma unroll 8
    for (int d = 0; d < D; ++d) {
        float x = w[(size_t)d * C + c];
        s = fmaf(x, x, s);
    }
    invn[c] = rsqrtf(fmaxf(s, EPS));
}

// ---------------------------------------------------------------------------
// Kernel 2: one wave (32 lanes) per row b.
//   ne[b,:]  = l2_normalize(embds[b,:])
//   gw[b,:]  = w[:,labels[b]] * invn[labels[b]]      (gathered normalized column)
//   sb[b]    = dot(ne[b,:], gw[b,:])  ( == cos_t[b, labels[b]] )
// ---------------------------------------------------------------------------
__global__ __launch_bounds__(32) void prep_kernel(const float* __restrict__ embds,
                                                  const float* __restrict__ w,
                                                  const int* __restrict__ labels,
                                                  const float* __restrict__ invn,
                                                  float* __restrict__ ne,
                                                  float* __restrict__ gw,
                                                  float* __restrict__ sb) {
    const int b = blockIdx.x;
    const int t = threadIdx.x;  // 0..31, 4 elements each (D=128)

    const float4 e = ((const float4*)(embds + (size_t)b * D))[t];
    float ss = e.x * e.x + e.y * e.y + e.z * e.z + e.w * e.w;
#pragma unroll
    for (int off = 16; off > 0; off >>= 1) ss += __shfl_xor(ss, off, 32);
    const float inv = rsqrtf(fmaxf(ss, EPS));

    const float4 nev = make_float4(e.x * inv, e.y * inv, e.z * inv, e.w * inv);
    ((float4*)(ne + (size_t)b * D))[t] = nev;

    const int lab = labels[b];
    const float wn = invn[lab];
    const float g0 = w[(size_t)(4 * t + 0) * C + lab] * wn;
    const float g1 = w[(size_t)(4 * t + 1) * C + lab] * wn;
    const float g2 = w[(size_t)(4 * t + 2) * C + lab] * wn;
    const float g3 = w[(size_t)(4 * t + 3) * C + lab] * wn;
    ((float4*)(gw + (size_t)b * D))[t] = make_float4(g0, g1, g2, g3);

    float dot = nev.x * g0 + nev.y * g1 + nev.z * g2 + nev.w * g3;
#pragma unroll
    for (int off = 16; off > 0; off >>= 1) dot += __shfl_xor(dot, off, 32);
    if (t == 0) sb[b] = dot;
}

// ---------------------------------------------------------------------------
// Kernel 3: dual WMMA f32 GEMM + fused epilogue.
//   Wave footprint: 2 b-tiles x 2 c-tiles (32 b rows x 32 c cols).
//   Per k-step: 4 A b64 loads + 4 B b32 loads feed 8 v_wmma_f32_16x16x4_f32
//   (1 load per WMMA; both A and B fragments are reused).
//   accC = ne x w-tile   (-> cos_t after scaling by invn[c])
//   accW = gw x w-tile   (-> wdot  after scaling by invn[c])
//   win  = (sb[b] - cos) * rsqrt(max(2 - 2*wdot, eps))
//   pen  = MARGIN - min(MARGIN, win)
//   cos_t is written into the logits slot of d_out (softmax'd in-place later).
//
// Fragment layouts per CDNA5 ISA (V_WMMA_F32_16X16X4_F32, wave32):
//   A 16x4 : lane L -> m = L%16 ; VGPR0/1 hold K = 2*(L/16) + {0,1}
//   B 4x16 : lane L -> n = L%16 ; VGPR0/1 hold K = 2*(L/16) + {0,1}
//   C 16x16: lane L -> n = L%16 ; VGPR r holds m = 8*(L/16) + r
// ---------------------------------------------------------------------------
__global__ __launch_bounds__(64) void gemm_kernel(const float* __restrict__ w,
                                                  const float* __restrict__ ne,
                                                  const float* __restrict__ gw,
                                                  const float* __restrict__ invn,
                                                  const float* __restrict__ sb,
                                                  float* __restrict__ cos_out,   // d_out[0 .. B*C)
                                                  float* __restrict__ pen_out) { // d_out[B*C .. 2*B*C)
    const int wave = threadIdx.x >> 5;   // 0..1
    const int lane = threadIdx.x & 31;
    const int half = lane >> 4;          // 0 or 1
    const int lm   = lane & 15;          // 0..15

    const int c0 = blockIdx.x * 64 + wave * 32;  // 125 blocks * 64 = 8000 exactly
    const int bb = blockIdx.y * 32;              // 8 * 32 = 256 = B exactly

    v8f accC[2][2], accW[2][2];  // [b-sub][c-sub]
#pragma unroll
    for (int sbt = 0; sbt < 2; ++sbt)
#pragma unroll
        for (int sc = 0; sc < 2; ++sc) { accC[sbt][sc] = (v8f){}; accW[sbt][sc] = (v8f){}; }

    // K = 128 -> 32 steps of K=4
    for (int j = 0; j < 32; ++j) {
        const int kb = 4 * j + 2 * half;

        v2f bf[2];
#pragma unroll
        for (int sc = 0; sc < 2; ++sc) {
            bf[sc].x = w[(size_t)kb * C + c0 + sc * 16 + lm];
            bf[sc].y = w[(size_t)(kb + 1) * C + c0 + sc * 16 + lm];
        }

#pragma unroll
        for (int sbt = 0; sbt < 2; ++sbt) {
            const int m = bb + sbt * 16 + lm;
            const v2f an = *(const v2f*)(ne + (size_t)m * D + kb);
            const v2f ag = *(const v2f*)(gw + (size_t)m * D + kb);
#pragma unroll
            for (int sc = 0; sc < 2; ++sc) {
                accC[sbt][sc] = __builtin_amdgcn_wmma_f32_16x16x4_f32(
                    false, an, false, bf[sc], (short)0, accC[sbt][sc], false, false);
                accW[sbt][sc] = __builtin_amdgcn_wmma_f32_16x16x4_f32(
                    false, ag, false, bf[sc], (short)0, accW[sbt][sc], false, false);
            }
        }
    }

    // Epilogue
#pragma unroll
    for (int sc = 0; sc < 2; ++sc) {
        const int   c  = c0 + sc * 16 + lm;
        const float wn = invn[c];
#pragma unroll
        for (int sbt = 0; sbt < 2; ++sbt) {
#pragma unroll
            for (int r = 0; r < 8; ++r) {
                const int   b    = bb + sbt * 16 + half * 8 + r;
                const float cosv = accC[sbt][sc][r] * wn;
                const float wd   = accW[sbt][sc][r] * wn;
                const float win  = (sb[b] - cosv) * rsqrtf(fmaxf(2.f - 2.f * wd, EPS));
                const float pen  = MARGIN - fminf(MARGIN, win);
                cos_out[(size_t)b * C + c] = cosv;
                pen_out[(size_t)b * C + c] = pen;
            }
        }
    }
}

// ---------------------------------------------------------------------------
// Kernel 4: in-place row softmax of SCALE*cos over C=8000, one block per row.
// ---------------------------------------------------------------------------
__global__ __launch_bounds__(256) void softmax_kernel(float* __restrict__ io) {
    __shared__ float red[256];
    const int b = blockIdx.x;
    const int t = threadIdx.x;
    float* row = io + (size_t)b * C;

    // Pass 1: max
    float mx = -INFINITY;
    for (int c = t; c < C; c += 256) mx = fmaxf(mx, row[c]);
    red[t] = mx;
    __syncthreads();
#pragma unroll
    for (int s = 128; s > 0; s >>= 1) {
        if (t < s) red[t] = fmaxf(red[t], red[t + s]);
        __syncthreads();
    }
    const float m = red[0] * SCALE;
    __syncthreads();

    // Pass 2: sum of exp
    float sum = 0.f;
    for (int c = t; c < C; c += 256) sum += expf(SCALE * row[c] - m);
    red[t] = sum;
    __syncthreads();
#pragma unroll
    for (int s = 128; s > 0; s >>= 1) {
        if (t < s) red[t] += red[t + s];
        __syncthreads();
    }
    const float inv = 1.f / red[0];
    __syncthreads();

    // Pass 3: write logits in place
    for (int c = t; c < C; c += 256) row[c] = expf(SCALE * row[c] - m) * inv;
}

// ---------------------------------------------------------------------------
// Launch
// ---------------------------------------------------------------------------
extern "C" void kernel_launch(void* const* d_in, const int* in_sizes, int n_in,
                              void* d_out, int out_size, void* d_ws, size_t ws_size,
                              hipStream_t stream) {
    const float* embds  = (const float*)d_in[0];   // [B, D]
    const float* w      = (const float*)d_in[1];   // [D, C]
    const int*   labels = (const int*)d_in[2];     // [B]

    float* out    = (float*)d_out;
    float* logits = out;                     // [B, C]  (holds cos_t, then softmax in place)
    float* pens   = out + (size_t)B * C;     // [B, C]

    // Workspace layout (floats), ~296 KB total:
    float* ws   = (float*)d_ws;
    float* invn = ws;          // [C]      = 8000
    float* ne   = ws + 8192;   // [B, D]   = 32768
    float* gw   = ws + 40960;  // [B, D]   = 32768
    float* sbuf = ws + 73728;  // [B]      = 256

    wnorm_kernel<<<dim3((C + 255) / 256), dim3(256), 0, stream>>>(w, invn);
    prep_kernel<<<dim3(B), dim3(32), 0, stream>>>(embds, w, labels, invn, ne, gw, sbuf);
    gemm_kernel<<<dim3(C / 64, B / 32), dim3(64), 0, stream>>>(w, ne, gw, invn, sbuf,
                                                               logits, pens);
    softmax_kernel<<<dim3(B), dim3(256), 0, stream>>>(logits);
}